// GraphModuleV5_46943992546025
// MI455X (gfx1250) — compile-verified
//
#include <hip/hip_runtime.h>
#include <math.h>
#include <stdint.h>

typedef __attribute__((ext_vector_type(2))) float v2f;
typedef __attribute__((ext_vector_type(4))) float v4f;
typedef __attribute__((ext_vector_type(8))) float v8f;

#define D_FEAT   256
#define H_HEADS  10
#define HPAD     16
#define TILE_N   16
#define F_STRIDE 264   // row stride (floats); 264*4 % 16 == 0 (b128-aligned), numerator-B conflict-free
#define W_STRIDE 20    // padded stride for 16-wide tiles; keeps ds ops conflict-free
#define NTHREADS 512   // 16 waves of 32

// Issue async DMA of one 16x256 feature tile into an LDS buffer.
// Tail rows (>= m) are zero-filled with plain ds stores (covered by dscnt at the
// next __syncthreads); live rows go through GLOBAL_LOAD_ASYNC_TO_LDS_B128
// (ASYNCcnt), EXEC-masked per lane.
__device__ __forceinline__ void issue_tile(const float* __restrict__ features,
                                           int n0, int m, int t,
                                           float (*Fb)[F_STRIDE])
{
    for (int v = t; v < TILE_N * (D_FEAT / 4); v += NTHREADS) {
        const int row = v >> 6;          // D_FEAT/4 == 64
        const int c4  = v & 63;
        if (row < m) {
            unsigned long long gaddr =
                (unsigned long long)(uintptr_t)(features + (size_t)(n0 + row) * D_FEAT + c4 * 4);
            unsigned int laddr = (unsigned int)(uintptr_t)&Fb[row][c4 * 4];  // low 32 bits = LDS offset
            asm volatile("global_load_async_to_lds_b128 %0, %1, off th:TH_LOAD_NT"
                         :: "v"(laddr), "v"(gaddr) : "memory");
        } else {
            v4f z = {};
            *(v4f*)&Fb[row][c4 * 4] = z;
        }
    }
}

__global__ __launch_bounds__(NTHREADS)
void seg_weighted_mean_kernel(const float* __restrict__ features,
                              const float* __restrict__ W,
                              const float* __restrict__ b,
                              const int*   __restrict__ seg,
                              float*       __restrict__ out,
                              int N)
{
    __shared__ float F_lds[2][TILE_N][F_STRIDE]; // double-buffered feature tiles [node][d]
    __shared__ float W_lds[D_FEAT][W_STRIDE];    // weight matrix [d][head(pad16)]
    __shared__ float L_lds[TILE_N][W_STRIDE];    // logits accum [node][head]
    __shared__ float wt_lds[TILE_N][W_STRIDE];   // sigmoid weights [node][head]
    __shared__ float b_lds[HPAD];
    __shared__ float denom_lds[HPAD];

    const int t    = threadIdx.x;
    const int g    = blockIdx.x;
    const int wave = t >> 5;
    const int lane = t & 31;
    const int half = lane >> 4;   // 0: lanes 0-15, 1: lanes 16-31
    const int l16  = lane & 15;
    const int dsl  = wave * 16;   // this wave's d-slice

    // ---- preload W (heads padded to 16 with zeros) and bias ----
    for (int idx = t; idx < D_FEAT * HPAD; idx += NTHREADS) {
        int k = idx >> 4, h = idx & 15;
        W_lds[k][h] = (h < H_HEADS) ? W[k * H_HEADS + h] : 0.0f;
    }
    if (t < HPAD) b_lds[t] = (t < H_HEADS) ? b[t] : 0.0f;

    // ---- uniform binary search for [start, end) of this segment ----
    int lo = 0, hi = N;
    while (lo < hi) { int mid = (lo + hi) >> 1; if (seg[mid] < g) lo = mid + 1; else hi = mid; }
    const int start = lo;
    hi = N;
    while (lo < hi) { int mid = (lo + hi) >> 1; if (seg[mid] <= g) lo = mid + 1; else hi = mid; }
    const int end = lo;

    const int nchunks = (end - start + TILE_N - 1) / TILE_N;

    v8f   accN = {};     // numerator tile: [16 heads x 16 d], row r -> head r (+8 for upper half)
    float dacc = 0.0f;   // denominator, owned by threads 0..15 (head == t)

    // ---- prologue: start DMA of first tile into buffer 0 ----
    if (nchunks > 0) {
        int m0 = end - start; if (m0 > TILE_N) m0 = TILE_N;
        issue_tile(features, start, m0, t, F_lds[0]);
    }

    int p = 0;
    for (int c = 0; c < nchunks; ++c, p ^= 1) {
        const int n0 = start + c * TILE_N;
        int m = end - n0; if (m > TILE_N) m = TILE_N;

        // ---- step A: wait own DMA for buf[p]; zero logits accum; prefetch next tile ----
        asm volatile("s_wait_asynccnt 0x0" ::: "memory");
        for (int v = t; v < TILE_N * W_STRIDE; v += NTHREADS)
            (&L_lds[0][0])[v] = 0.0f;
        if (c + 1 < nchunks) {
            const int n1 = n0 + TILE_N;
            int m1 = end - n1; if (m1 > TILE_N) m1 = TILE_N;
            issue_tile(features, n1, m1, t, F_lds[p ^ 1]);   // other buffer: readers done at prev Bar4
        }
        __syncthreads();                                     // Bar1: tiles visible, L zeroed

        // ---- phase B: logits = F(16x256) @ Wpad(256x16), split-K across 16 waves ----
        {
            v8f cacc = {};
            const int ks = wave * 16;
            #pragma unroll
            for (int i = 0; i < 4; ++i) {
                const int kb = ks + i * 4 + 2 * half;        // K pair base for this half-wave
                v2f a, bb;
                a.x  = F_lds[p][l16][kb];                    // A: M=node(l16), K
                a.y  = F_lds[p][l16][kb + 1];
                bb.x = W_lds[kb][l16];                       // B: K, N=head(l16)
                bb.y = W_lds[kb + 1][l16];
                cacc = __builtin_amdgcn_wmma_f32_16x16x4_f32(false, a, false, bb,
                                                             (short)0, cacc, false, false);
            }
            #pragma unroll
            for (int r = 0; r < 8; ++r)                      // reduce split-K partials in LDS
                atomicAdd(&L_lds[r + 8 * half][l16], cacc[r]);
        }
        __syncthreads();                                     // Bar2

        // ---- phase C: sigmoid + mask (tail nodes & padded heads -> weight 0) ----
        if (t < TILE_N * HPAD) {
            const int n = t >> 4, h = t & 15;
            const float x = L_lds[n][h] + b_lds[h];
            wt_lds[n][h] = (n < m && h < H_HEADS) ? (1.0f / (1.0f + __expf(-x))) : 0.0f;
        }
        __syncthreads();                                     // Bar3

        // denominator accumulation (threads 0..15 own one head each)
        if (t < HPAD) {
            float s = 0.0f;
            #pragma unroll
            for (int n = 0; n < TILE_N; ++n) s += wt_lds[n][t];
            dacc += s;
        }

        // ---- phase D: numerator tile += Wt^T(16x16) @ F(16x16-slice), fp32 WMMA ----
        #pragma unroll
        for (int i = 0; i < 4; ++i) {
            const int nb = i * 4 + 2 * half;                 // K = node pair base
            v2f a, bb;
            a.x  = wt_lds[nb][l16];                          // A: M=head(l16), K=node
            a.y  = wt_lds[nb + 1][l16];
            bb.x = F_lds[p][nb][dsl + l16];                  // B: K=node, N=d
            bb.y = F_lds[p][nb + 1][dsl + l16];
            accN = __builtin_amdgcn_wmma_f32_16x16x4_f32(false, a, false, bb,
                                                         (short)0, accN, false, false);
        }
        __syncthreads();                                     // Bar4: frees wt_lds and buf[p]
    }

    if (t < HPAD) denom_lds[t] = dacc;
    __syncthreads();

    // ---- epilogue: keys = num / denom ; rows M=10..15 are padding, dropped ----
    #pragma unroll
    for (int r = 0; r < 8; ++r) {
        const int h = r + 8 * half;
        if (h < H_HEADS) {
            const float dnm = denom_lds[h];
            out[(size_t)g * (H_HEADS * D_FEAT) + (size_t)h * D_FEAT + dsl + l16] = accN[r] / dnm;
        }
    }
}

extern "C" void kernel_launch(void* const* d_in, const int* in_sizes, int n_in,
                              void* d_out, int out_size, void* d_ws, size_t ws_size,
                              hipStream_t stream) {
    const float* features = (const float*)d_in[0];
    const float* W        = (const float*)d_in[1];
    const float* b        = (const float*)d_in[2];
    const int*   seg      = (const int*)d_in[3];
    const int N = in_sizes[3];                       // 500,000
    const int G = out_size / (H_HEADS * D_FEAT);     // 8192

    dim3 grid(G), block(NTHREADS);
    hipLaunchKernelGGL(seg_weighted_mean_kernel, grid, block, 0, stream,
                       features, W, b, seg, (float*)d_out, N);
}